// RANSAC_10024453669067
// MI455X (gfx1250) — compile-verified
//
#include <hip/hip_runtime.h>
#include <hip/hip_bf16.h>
#include <math.h>

typedef __attribute__((ext_vector_type(2))) float v2f;
typedef __attribute__((ext_vector_type(8))) float v8f;

#define EPS_     1e-8f
#define INL_TH_  2.0f
#define ROOT2_   1.41421356237f
#define NTH_LO   256
#define NWAVES_LO 8

// ---------------------------------------------------------------- helpers

__device__ __forceinline__ float xfer_err_sq(const float* H, float x, float y,
                                             float x2, float y2) {
  float px = H[0] * x + H[1] * y + H[2];
  float py = H[3] * x + H[4] * y + H[5];
  float pz = H[6] * x + H[7] * y + H[8];
  float s  = (fabsf(pz) > EPS_) ? (1.0f / pz) : 1.0f;
  float dx = px * s - x2;
  float dy = py * s - y2;
  return dx * dx + dy * dy;
}

// Smallest eigenvector of packed-upper symmetric 9x9 AtA (Cholesky + inverse
// power iteration), then denormalize: H = inv(T2) * Hn * T1, H /= H[2][2]+eps.
__device__ void solve_h_from_ata(const float* Sp, float s1, float mx1, float my1,
                                 float s2, float mx2, float my2, float* Hout) {
  float A[9][9];
  int k = 0;
  for (int i = 0; i < 9; ++i)
    for (int j = i; j < 9; ++j) {
      float v = Sp[k++];
      A[i][j] = v;
      A[j][i] = v;
    }
  float tr = 0.f;
  for (int i = 0; i < 9; ++i) tr += A[i][i];
  float ridge = tr * 1e-8f + 1e-18f;
  for (int i = 0; i < 9; ++i) A[i][i] += ridge;

  // in-place Cholesky (lower triangle of A becomes L)
  for (int j = 0; j < 9; ++j) {
    float d = A[j][j];
    for (int kk = 0; kk < j; ++kk) d -= A[j][kk] * A[j][kk];
    d = sqrtf(fmaxf(d, 1e-30f));
    A[j][j] = d;
    float inv = 1.0f / d;
    for (int i = j + 1; i < 9; ++i) {
      float s = A[i][j];
      for (int kk = 0; kk < j; ++kk) s -= A[i][kk] * A[j][kk];
      A[i][j] = s * inv;
    }
  }

  // inverse power iteration: x -> (L L^T)^-1 x, normalized
  float x[9];
  for (int i = 0; i < 9; ++i) x[i] = 0.3333333f + 0.011f * (float)i;
  for (int it = 0; it < 40; ++it) {
    float y[9];
    for (int i = 0; i < 9; ++i) {
      float s = x[i];
      for (int kk = 0; kk < i; ++kk) s -= A[i][kk] * y[kk];
      y[i] = s / A[i][i];
    }
    for (int i = 8; i >= 0; --i) {
      float s = y[i];
      for (int kk = i + 1; kk < 9; ++kk) s -= A[kk][i] * x[kk];
      x[i] = s / A[i][i];
    }
    float n2 = 0.f;
    for (int i = 0; i < 9; ++i) n2 += x[i] * x[i];
    float rn = rsqrtf(n2 + 1e-30f);
    for (int i = 0; i < 9; ++i) x[i] *= rn;
  }

  // Hn rows -> M = Hn * T1 ; H = inv(T2) * M
  float M[3][3];
  for (int r = 0; r < 3; ++r) {
    float h0 = x[r * 3 + 0], h1 = x[r * 3 + 1], h2 = x[r * 3 + 2];
    M[r][0] = s1 * h0;
    M[r][1] = s1 * h1;
    M[r][2] = h2 - s1 * (mx1 * h0 + my1 * h1);
  }
  float is2 = 1.0f / s2;
  float H[9];
  for (int c = 0; c < 3; ++c) {
    H[0 + c] = is2 * M[0][c] + mx2 * M[2][c];
    H[3 + c] = is2 * M[1][c] + my2 * M[2][c];
    H[6 + c] = M[2][c];
  }
  float idn = 1.0f / (H[8] + EPS_);
  for (int i = 0; i < 9; ++i) Hout[i] = H[i] * idn;
}

// ---------------------------------------------------------------- kernel 1
// One thread per hypothesis: 4-pt normalization + DLT AtA + eigen solve.
__global__ void hypo_kernel(const float* __restrict__ kp1,
                            const float* __restrict__ kp2,
                            const int* __restrict__ idxs,
                            float* __restrict__ models,
                            float* __restrict__ valid, int B) {
  int b = blockIdx.x * blockDim.x + threadIdx.x;
  if (b >= B) return;

  float X1[4], Y1[4], X2[4], Y2[4];
  for (int j = 0; j < 4; ++j) {
    int id = idxs[4 * b + j];
    X1[j] = kp1[2 * id];
    Y1[j] = kp1[2 * id + 1];
    X2[j] = kp2[2 * id];
    Y2[j] = kp2[2 * id + 1];
  }

  // sample validity: orientation signs of 4 triplets must match
  const int TA[4] = {0, 0, 0, 1};
  const int TB[4] = {1, 2, 1, 2};
  const int TC[4] = {2, 3, 3, 3};
  bool vs = true;
  for (int t = 0; t < 4; ++t) {
    int a = TA[t], bb = TB[t], c = TC[t];
    float d1 = (X1[bb] - X1[a]) * (Y1[c] - Y1[a]) - (X1[c] - X1[a]) * (Y1[bb] - Y1[a]);
    float d2 = (X2[bb] - X2[a]) * (Y2[c] - Y2[a]) - (X2[c] - X2[a]) * (Y2[bb] - Y2[a]);
    int sg1 = (d1 > 0.f) - (d1 < 0.f);
    int sg2 = (d2 > 0.f) - (d2 < 0.f);
    vs = vs && (sg1 == sg2);
  }

  // normalization
  float mx1 = 0.25f * (X1[0] + X1[1] + X1[2] + X1[3]);
  float my1 = 0.25f * (Y1[0] + Y1[1] + Y1[2] + Y1[3]);
  float mx2 = 0.25f * (X2[0] + X2[1] + X2[2] + X2[3]);
  float my2 = 0.25f * (Y2[0] + Y2[1] + Y2[2] + Y2[3]);
  float sc1 = 0.f, sc2 = 0.f;
  for (int j = 0; j < 4; ++j) {
    float dx = X1[j] - mx1, dy = Y1[j] - my1;
    sc1 += sqrtf(dx * dx + dy * dy);
    dx = X2[j] - mx2; dy = Y2[j] - my2;
    sc2 += sqrtf(dx * dx + dy * dy);
  }
  sc1 *= 0.25f; sc2 *= 0.25f;
  float s1 = ROOT2_ / (sc1 + EPS_);
  float s2 = ROOT2_ / (sc2 + EPS_);

  float S[45];
  for (int i = 0; i < 45; ++i) S[i] = 0.f;
  for (int j = 0; j < 4; ++j) {
    float x1 = (X1[j] - mx1) * s1, y1 = (Y1[j] - my1) * s1;
    float x2 = (X2[j] - mx2) * s2, y2 = (Y2[j] - my2) * s2;
    float ax[9] = {0.f, 0.f, 0.f, -x1, -y1, -1.f, y2 * x1, y2 * y1, y2};
    float ay[9] = {x1, y1, 1.f, 0.f, 0.f, 0.f, -x2 * x1, -x2 * y1, -x2};
    int k = 0;
    for (int i = 0; i < 9; ++i)
      for (int j2 = i; j2 < 9; ++j2)
        S[k++] += ax[i] * ax[j2] + ay[i] * ay[j2];
  }

  float H[9];
  solve_h_from_ata(S, s1, mx1, my1, s2, mx2, my2, H);

  float dm = fminf(fabsf(H[0]), fminf(fabsf(H[4]), fabsf(H[8])));
  valid[b] = (vs && (dm > 1e-4f)) ? 1.f : 0.f;
  for (int i = 0; i < 9; ++i) models[b * 9 + i] = H[i];
}

// ---------------------------------------------------------------- kernel 2
// 4 waves per block; each wave scores a group of 4 models. Point data is
// staged through LDS in 64-point chunks with GLOBAL_LOAD_ASYNC_TO_LDS_B128
// (ASYNCcnt), then consumed by WMMA F32 16x16x4:
//   A (16x4) rows 0-5  = model0/model1 H-rows,  rows 8-13 = model2/model3.
//   B (4x16) cols      = 16 points [x, y, 1, 0].
//   D (16x16): lane n (<16) holds rows 0..7 => models 0,1 of point n;
//              lane n+16 holds rows 8..15   => models 2,3 of point n.
__global__ void __launch_bounds__(128) score_kernel(
    const float* __restrict__ kp1, const float* __restrict__ kp2,
    const float* __restrict__ models, const float* __restrict__ valid,
    float* __restrict__ scores, int B, int N) {
  __shared__ float sP1[128];  // 64 points x {x,y}
  __shared__ float sP2[128];

  int tid  = threadIdx.x;
  int lane = tid & 31;
  int wib  = tid >> 5;                    // wave-in-block: 0..3
  int g    = blockIdx.x * 4 + wib;        // model group (4 models)
  int G    = (B + 3) >> 2;
  bool active = (g < G);

  int M  = lane & 15;          // A-matrix row / point slot
  int hi = lane >> 4;          // 0: K=0,1  1: K=2,3

  // A operand: H-row assignment for this lane's matrix row M
  int ml = -1, r = 0;
  if (M < 6)                 { ml = M / 3;           r = M % 3; }
  else if (M >= 8 && M < 14) { ml = 2 + (M - 8) / 3; r = (M - 8) % 3; }
  v2f a; a.x = 0.f; a.y = 0.f;
  if (active && ml >= 0) {
    const float* h = models + (size_t)(4 * g + ml) * 9 + r * 3;
    float rv0 = h[0], rv1 = h[1], rv2 = h[2];
    a.x = hi ? rv2 : rv0;   // K=2 : K=0
    a.y = hi ? 0.f : rv1;   // K=3 : K=1
  }

  // async-staging addresses (wave 0 stages the whole chunk: 32 lanes x 16B)
  unsigned int lds1 = ((unsigned int)(uintptr_t)sP1) + (unsigned int)lane * 16u;
  unsigned int lds2 = ((unsigned int)(uintptr_t)sP2) + (unsigned int)lane * 16u;
  unsigned long long base1 = (unsigned long long)(uintptr_t)kp1;
  unsigned long long base2 = (unsigned long long)(uintptr_t)kp2;
  unsigned int limit = 8u * (unsigned int)N - 16u;  // last aligned 16B chunk

  float cnt0 = 0.f, cnt1 = 0.f;
  int nChunks = (N + 63) / 64;
  for (int ch = 0; ch < nChunks; ++ch) {
    __syncthreads();  // previous chunk fully consumed before overwrite
    if (wib == 0) {
      unsigned int off = (unsigned int)ch * 512u + (unsigned int)lane * 16u;
      if (off > limit) off = limit;  // tail: clamp (dupes masked at use)
      unsigned long long ga1 = base1 + off;
      unsigned long long ga2 = base2 + off;
      asm volatile(
          "global_load_async_to_lds_b128 %0, %2, off\n\t"
          "global_load_async_to_lds_b128 %1, %3, off\n\t"
          "s_wait_asynccnt 0"
          :
          : "v"(lds1), "v"(lds2), "v"(ga1), "v"(ga2)
          : "memory");
    }
    __syncthreads();  // LDS chunk visible to all 4 waves

    for (int sub = 0; sub < 4; ++sub) {
      int p  = ch * 64 + sub * 16 + M;
      bool inb = (p < N);
      float2 q1 = ((const float2*)sP1)[sub * 16 + M];
      float2 q2 = ((const float2*)sP2)[sub * 16 + M];
      float x2 = q2.x, y2 = q2.y;

      v2f bm;
      bm.x = hi ? 1.0f : q1.x;  // K=2 (homog 1) : K=0 (x)
      bm.y = hi ? 0.0f : q1.y;  // K=3 (pad)     : K=1 (y)

      v8f c = {};
      v8f d = __builtin_amdgcn_wmma_f32_16x16x4_f32(
          false, a, false, bm, (short)0, c, false, false);

      {  // model pair A (rows 0..2 of this half)
        float pz = d[2];
        float s  = (fabsf(pz) > EPS_) ? (1.0f / pz) : 1.0f;
        float dx = d[0] * s - x2, dy = d[1] * s - y2;
        float e  = dx * dx + dy * dy;
        cnt0 += (inb && (e <= INL_TH_)) ? 1.f : 0.f;
      }
      {  // model pair B (rows 3..5 of this half)
        float pz = d[5];
        float s  = (fabsf(pz) > EPS_) ? (1.0f / pz) : 1.0f;
        float dx = d[3] * s - x2, dy = d[4] * s - y2;
        float e  = dx * dx + dy * dy;
        cnt1 += (inb && (e <= INL_TH_)) ? 1.f : 0.f;
      }
    }
  }

  // butterfly within each 16-lane half
  for (int o = 8; o >= 1; o >>= 1) {
    cnt0 += __shfl_xor(cnt0, o, 32);
    cnt1 += __shfl_xor(cnt1, o, 32);
  }
  if (active && M == 0) {  // lane 0 -> models 4g,4g+1 ; lane 16 -> 4g+2,4g+3
    int base = 4 * g + hi * 2;
    scores[base]     = (valid[base]     > 0.5f) ? cnt0 : -1.0f;
    scores[base + 1] = (valid[base + 1] > 0.5f) ? cnt1 : -1.0f;
  }
}

// ---------------------------------------------------------------- kernel 3
__global__ void argmax_kernel(const float* __restrict__ scores,
                              const float* __restrict__ models,
                              float* __restrict__ cur, float* __restrict__ misc,
                              int B) {
  __shared__ float sV[256];
  __shared__ int   sI[256];
  int tid = threadIdx.x;
  float bv = -1e30f;
  int   bi = 0;
  for (int b = tid; b < B; b += blockDim.x) {
    float s = scores[b];
    if (s > bv || (s == bv && b < bi)) { bv = s; bi = b; }
  }
  sV[tid] = bv; sI[tid] = bi;
  __syncthreads();
  for (int s = 128; s > 0; s >>= 1) {
    if (tid < s) {
      if (sV[tid + s] > sV[tid] ||
          (sV[tid + s] == sV[tid] && sI[tid + s] < sI[tid])) {
        sV[tid] = sV[tid + s];
        sI[tid] = sI[tid + s];
      }
    }
    __syncthreads();
  }
  if (tid == 0) {
    int b = sI[0];
    misc[0] = sV[0];                 // initial best score (accept test)
    ((int*)misc)[1] = b;
    for (int i = 0; i < 9; ++i) cur[i] = models[b * 9 + i];
  }
}

// ---------------------------------------------------------------- kernel 4
__device__ float block_sum256(float v, float* sPar, int tid) {
  int lane = tid & 31, wid = tid >> 5;
  for (int o = 16; o > 0; o >>= 1) v += __shfl_down(v, o, 32);
  __syncthreads();
  if (lane == 0) sPar[wid] = v;
  __syncthreads();
  float r = 0.f;
  for (int i = 0; i < NWAVES_LO; ++i) r += sPar[i];  // fixed order: deterministic
  return r;
}

__device__ void dlt_coop(const float* kp1, const float* kp2, const float* w,
                         int N, float s1, float mx1, float my1, float s2,
                         float mx2, float my2, float* sRed, float* sH, int tid) {
  float acc[45];
  for (int i = 0; i < 45; ++i) acc[i] = 0.f;
  for (int p = tid; p < N; p += NTH_LO) {
    float wp = w[p];
    float x1 = (kp1[2 * p] - mx1) * s1, y1 = (kp1[2 * p + 1] - my1) * s1;
    float x2 = (kp2[2 * p] - mx2) * s2, y2 = (kp2[2 * p + 1] - my2) * s2;
    float ax[9] = {0.f, 0.f, 0.f, -x1, -y1, -1.f, y2 * x1, y2 * y1, y2};
    float ay[9] = {x1, y1, 1.f, 0.f, 0.f, 0.f, -x2 * x1, -x2 * y1, -x2};
    int k = 0;
    for (int i = 0; i < 9; ++i)
      for (int j = i; j < 9; ++j)
        acc[k++] += wp * (ax[i] * ax[j] + ay[i] * ay[j]);
  }
  int lane = tid & 31, wid = tid >> 5;
  for (int k = 0; k < 45; ++k) {
    float v = acc[k];
    for (int o = 16; o > 0; o >>= 1) v += __shfl_down(v, o, 32);
    if (lane == 0) sRed[wid * 45 + k] = v;
  }
  __syncthreads();
  if (tid == 0) {
    float Sp[45];
    for (int k = 0; k < 45; ++k) {
      float v = 0.f;
      for (int wv = 0; wv < NWAVES_LO; ++wv) v += sRed[wv * 45 + k];
      Sp[k] = v;
    }
    solve_h_from_ata(Sp, s1, mx1, my1, s2, mx2, my2, sH);
  }
  __syncthreads();
}

__global__ void __launch_bounds__(NTH_LO) lo_kernel(
    const float* __restrict__ kp1, const float* __restrict__ kp2,
    const float* __restrict__ cur, const float* __restrict__ misc,
    float* __restrict__ wbuf, float* __restrict__ inl, float* __restrict__ ilt,
    float* __restrict__ out, int N) {
  __shared__ float sRed[NWAVES_LO * 45];
  __shared__ float sH[9];
  __shared__ float sM[9];
  __shared__ float sPar[NWAVES_LO];
  __shared__ float sCtl[2];  // [0]=cur score, [1]=better flag
  int tid = threadIdx.x;

  // ---- full-set normalization (T1, T2 fixed for every polish iteration)
  float ax1 = 0.f, ay1 = 0.f, ax2 = 0.f, ay2 = 0.f;
  for (int p = tid; p < N; p += NTH_LO) {
    ax1 += kp1[2 * p]; ay1 += kp1[2 * p + 1];
    ax2 += kp2[2 * p]; ay2 += kp2[2 * p + 1];
  }
  float invN = 1.0f / (float)N;
  float mx1 = block_sum256(ax1, sPar, tid) * invN;
  float my1 = block_sum256(ay1, sPar, tid) * invN;
  float mx2 = block_sum256(ax2, sPar, tid) * invN;
  float my2 = block_sum256(ay2, sPar, tid) * invN;
  float a1 = 0.f, a2 = 0.f;
  for (int p = tid; p < N; p += NTH_LO) {
    float dx = kp1[2 * p] - mx1, dy = kp1[2 * p + 1] - my1;
    a1 += sqrtf(dx * dx + dy * dy);
    dx = kp2[2 * p] - mx2; dy = kp2[2 * p + 1] - my2;
    a2 += sqrtf(dx * dx + dy * dy);
  }
  float s1 = ROOT2_ / (block_sum256(a1, sPar, tid) * invN + EPS_);
  float s2 = ROOT2_ / (block_sum256(a2, sPar, tid) * invN + EPS_);

  // ---- init current best model + inliers
  if (tid == 0) {
    for (int i = 0; i < 9; ++i) sM[i] = cur[i];
    sCtl[0] = misc[0];
  }
  __syncthreads();
  for (int p = tid; p < N; p += NTH_LO) {
    float e = xfer_err_sq(sM, kp1[2 * p], kp1[2 * p + 1], kp2[2 * p], kp2[2 * p + 1]);
    inl[p] = (e <= INL_TH_) ? 1.f : 0.f;
  }
  __syncthreads();

  // ---- MAX_LO_ITERS = 5, each: polish (DLT_ITERS=5 weighted DLT), keep if better
  for (int lo = 0; lo < 5; ++lo) {
    dlt_coop(kp1, kp2, inl, N, s1, mx1, my1, s2, mx2, my2, sRed, sH, tid);
    for (int d = 0; d < 4; ++d) {
      for (int p = tid; p < N; p += NTH_LO) {
        float e = sqrtf(fmaxf(
            xfer_err_sq(sH, kp1[2 * p], kp1[2 * p + 1], kp2[2 * p], kp2[2 * p + 1]),
            0.f));
        wbuf[p] = inl[p] * __expf(-e / 18.0f);  // 2*SOFT_INL_TH^2 = 18
      }
      __syncthreads();
      dlt_coop(kp1, kp2, wbuf, N, s1, mx1, my1, s2, mx2, my2, sRed, sH, tid);
    }
    // evaluate polished model
    float c = 0.f;
    for (int p = tid; p < N; p += NTH_LO) {
      float e = xfer_err_sq(sH, kp1[2 * p], kp1[2 * p + 1], kp2[2 * p], kp2[2 * p + 1]);
      float ii = (e <= INL_TH_) ? 1.f : 0.f;
      ilt[p] = ii;
      c += ii;
    }
    __syncthreads();
    float scLo = block_sum256(c, sPar, tid);
    if (tid == 0) {
      if (scLo > sCtl[0]) {
        sCtl[0] = scLo;
        sCtl[1] = 1.f;
        for (int i = 0; i < 9; ++i) sM[i] = sH[i];
      } else {
        sCtl[1] = 0.f;
      }
    }
    __syncthreads();
    if (sCtl[1] > 0.5f) {
      for (int p = tid; p < N; p += NTH_LO) inl[p] = ilt[p];
    }
    __syncthreads();
  }

  // ---- output: accept iff initial best score > MIN_SAMPLE
  float accept = (misc[0] > 4.0f) ? 1.f : 0.f;
  if (tid < 9) out[tid] = accept * sM[tid];
  for (int p = tid; p < N; p += NTH_LO) out[9 + p] = accept * inl[p];
}

// ---------------------------------------------------------------- launch

extern "C" void kernel_launch(void* const* d_in, const int* in_sizes, int n_in,
                              void* d_out, int out_size, void* d_ws,
                              size_t ws_size, hipStream_t stream) {
  (void)n_in; (void)out_size; (void)ws_size;
  const float* kp1 = (const float*)d_in[0];
  const float* kp2 = (const float*)d_in[1];
  const int* idxs  = (const int*)d_in[2];
  int N = in_sizes[0] / 2;
  int B = in_sizes[2] / 4;
  float* out = (float*)d_out;
  float* wsf = (float*)d_ws;

  size_t off = 0;
  float* ws_models = wsf + off; off += (size_t)B * 9;
  float* ws_valid  = wsf + off; off += (size_t)B;
  float* ws_scores = wsf + off; off += (size_t)B;
  float* ws_cur    = wsf + off; off += 16;
  float* ws_misc   = wsf + off; off += 16;
  float* ws_w      = wsf + off; off += (size_t)N;
  float* ws_inl    = wsf + off; off += (size_t)N;
  float* ws_ilt    = wsf + off; off += (size_t)N;

  hypo_kernel<<<(B + 63) / 64, 64, 0, stream>>>(kp1, kp2, idxs, ws_models,
                                                ws_valid, B);

  int groups = (B + 3) / 4;               // 4 models per wave, 4 waves per block
  int blocks = (groups + 3) / 4;
  score_kernel<<<blocks, 128, 0, stream>>>(kp1, kp2, ws_models, ws_valid,
                                           ws_scores, B, N);

  argmax_kernel<<<1, 256, 0, stream>>>(ws_scores, ws_models, ws_cur, ws_misc, B);

  lo_kernel<<<1, NTH_LO, 0, stream>>>(kp1, kp2, ws_cur, ws_misc, ws_w, ws_inl,
                                      ws_ilt, out, N);
}